// HashingModel_35364760715522
// MI455X (gfx1250) — compile-verified
//
#include <hip/hip_runtime.h>
#include <hip/hip_bf16.h>

// ---------------------------------------------------------------------------
// Model dims (compile-time)
// ---------------------------------------------------------------------------
#define B     2048
#define LI    50
#define LT    77
#define FEAT  512
#define DM    1024   // d_model
#define DFF   2048
#define HID   4096
#define NBITS 64

typedef float v2f __attribute__((ext_vector_type(2)));
typedef float v8f __attribute__((ext_vector_type(8)));

// ---------------------------------------------------------------------------
// wave32 reductions
// ---------------------------------------------------------------------------
__device__ __forceinline__ float wave_sum(float v) {
#pragma unroll
  for (int off = 16; off > 0; off >>= 1) v += __shfl_xor(v, off, 32);
  return v;
}

__device__ __forceinline__ float block_sum256(float v, float* sbuf /*>=8*/) {
  __syncthreads();                      // protect sbuf reuse
  v = wave_sum(v);
  int w = threadIdx.x >> 5;
  if ((threadIdx.x & 31) == 0) sbuf[w] = v;
  __syncthreads();
  float r = (threadIdx.x < 8) ? sbuf[threadIdx.x] : 0.0f;
  if (threadIdx.x < 32) r = wave_sum(r);
  if (threadIdx.x == 0) sbuf[0] = r;
  __syncthreads();
  return sbuf[0];
}

// ---------------------------------------------------------------------------
// GEMM: C[M,N] = act( A[M,K] * W[N,K]^T + bias[N] )
// "NT" layout: both operands K-contiguous -> float2 WMMA fragment loads.
// Block = 256 thr = 8 waves as 2(M) x 4(N); block tile 32x64; K staged by 64.
// Tiles staged via GLOBAL_LOAD_ASYNC_TO_LDS_B128 (ASYNCcnt) into a ping-pong
// LDS double buffer so the DMA of slab k+1 overlaps the WMMAs of slab k.
// Math uses V_WMMA_F32_16X16X4_F32 (exact fp32 matrix MACs).
// ---------------------------------------------------------------------------
#define TM 32
#define TN 64
#define TK 64
#define LDSK (TK + 4)

__device__ __forceinline__ void async_b128_to_lds(const float* gptr,
                                                  const float* lptr) {
  unsigned lds = (unsigned)(size_t)lptr;  // AS3 offset = low 32 bits
  asm volatile("global_load_async_to_lds_b128 %0, %1, off"
               :: "v"(lds), "v"(gptr) : "memory");
}

__global__ __launch_bounds__(256) void gemm_nt_wmma_f32(
    const float* __restrict__ A, const float* __restrict__ W,
    const float* __restrict__ bias, float* __restrict__ C,
    int M, int N, int K, int act /*0=none,1=relu,2=tanh*/) {
  __shared__ float As[2][TM][LDSK];
  __shared__ float Bs[2][TN][LDSK];

  const int bm = blockIdx.y * TM;
  const int bn = blockIdx.x * TN;
  const int tid = threadIdx.x;
  const int wave = tid >> 5;
  const int lane = tid & 31;
  const int wm = (wave >> 2) << 4;   // 0 / 16
  const int wn = (wave & 3) << 4;    // 0 / 16 / 32 / 48

  v8f acc = {};

  const int mrow = wm + (lane & 15);
  const int nrow = wn + (lane & 15);
  const int koff = (lane >> 4) << 1;  // 0 or 2

  // issue one K-slab of async global->LDS DMA (no VGPR round-trip)
  auto stage = [&](int buf, int kb) {
#pragma unroll
    for (int i = tid; i < TM * TK / 4; i += 256) {   // A: 2 per thread
      int r = i / (TK / 4), c4 = i % (TK / 4);
      async_b128_to_lds(&A[(size_t)(bm + r) * K + kb + c4 * 4],
                        &As[buf][r][c4 * 4]);
    }
#pragma unroll
    for (int i = tid; i < TN * TK / 4; i += 256) {   // W: 4 per thread
      int r = i / (TK / 4), c4 = i % (TK / 4);
      async_b128_to_lds(&W[(size_t)(bn + r) * K + kb + c4 * 4],
                        &Bs[buf][r][c4 * 4]);
    }
  };

  stage(0, 0);
  asm volatile("s_wait_asynccnt 0x0" ::: "memory");
  __syncthreads();

  int buf = 0;
  for (int kb = 0; kb < K; kb += TK, buf ^= 1) {
    if (kb + TK < K) stage(buf ^ 1, kb + TK);   // prefetch next slab (async)
#pragma unroll
    for (int kk = 0; kk < TK; kk += 4) {
      v2f a = *(const v2f*)&As[buf][mrow][kk + koff];
      v2f b = *(const v2f*)&Bs[buf][nrow][kk + koff];
      acc = __builtin_amdgcn_wmma_f32_16x16x4_f32(
          /*neg_a=*/false, a, /*neg_b=*/false, b,
          /*c_mod=*/(short)0, acc, /*reuse_a=*/false, /*reuse_b=*/false);
    }
    asm volatile("s_wait_asynccnt 0x0" ::: "memory");
    __syncthreads();
  }

  // C/D layout: VGPR r -> (M = wm + r + 8*(lane>=16), N = wn + lane%16)
  const int n = bn + wn + (lane & 15);
  const float bv = bias ? bias[n] : 0.0f;
  const int mbase = bm + wm + ((lane >> 4) << 3);
#pragma unroll
  for (int r = 0; r < 8; ++r) {
    float v = acc[r] + bv;
    if (act == 1) v = fmaxf(v, 0.0f);
    else if (act == 2) v = tanhf(v);
    C[(size_t)(mbase + r) * N + n] = v;
  }
}

// ---------------------------------------------------------------------------
// Pooling stage: one block per sample. Computes (per b):
//   cap_mean (masked), img_mean, aggr[i]=img_i . cap_mean (== img_w),
//   txt_w[j]=cap_j . img_mean, top-30 mean img_k, weighted pools img_tm/txt_tm.
// Writes concat(img_k, cap_mean) -> X0[b,1024], concat(img_tm, txt_tm) -> X1.
// img_tokens: [LI,B,512] seq-first; txt_tokens: [LT,B,512]; mask: [B,LT] bool.
// ---------------------------------------------------------------------------
__global__ __launch_bounds__(256) void pool_kernel(
    const float* __restrict__ img, const float* __restrict__ txt,
    const unsigned char* __restrict__ kpm, float* __restrict__ X0,
    float* __restrict__ X1) {
  __shared__ float s_capmean[FEAT];
  __shared__ float s_imgmean[FEAT];
  __shared__ float s_aggr[64];
  __shared__ float s_txtw[96];
  __shared__ int s_sel[64];
  __shared__ float s_sc[2];

  const int b = blockIdx.x;
  const int t = threadIdx.x;
  const int wave = t >> 5, lane = t & 31;

  if (t < 64) { s_sel[t] = 0; s_aggr[t] = 0.0f; }
  if (t < 96) s_txtw[t] = 0.0f;

  // phase A: means (thread owns channels t and t+256)
  float i0 = 0.f, i1 = 0.f;
  for (int i = 0; i < LI; ++i) {
    const float* r = &img[((size_t)i * B + b) * FEAT];
    i0 += r[t]; i1 += r[t + 256];
  }
  float c0 = 0.f, c1 = 0.f;
  for (int j = 0; j < LT; ++j) {
    if (!kpm[(size_t)b * LT + j]) {
      const float* r = &txt[((size_t)j * B + b) * FEAT];
      c0 += r[t]; c1 += r[t + 256];
    }
  }
  s_imgmean[t] = i0 / (float)LI;  s_imgmean[t + 256] = i1 / (float)LI;
  s_capmean[t] = c0 / (float)LT;  s_capmean[t + 256] = c1 / (float)LT;
  __syncthreads();

  // phase B: dot products (wave w handles rows w, w+8, ...)
  for (int i = wave; i < LI; i += 8) {
    const float* r = &img[((size_t)i * B + b) * FEAT];
    float p = 0.f;
    for (int d = lane; d < FEAT; d += 32) p += r[d] * s_capmean[d];
    p = wave_sum(p);
    if (lane == 0) s_aggr[i] = p;
  }
  for (int j = wave; j < LT; j += 8) {
    float p = 0.f;
    if (!kpm[(size_t)b * LT + j]) {
      const float* r = &txt[((size_t)j * B + b) * FEAT];
      for (int d = lane; d < FEAT; d += 32) p += r[d] * s_imgmean[d];
    }
    p = wave_sum(p);
    if (lane == 0) s_txtw[j] = p;
  }
  __syncthreads();

  // phase C: sums + top-30 selection (tiny, serial on thread 0)
  if (t == 0) {
    float ws = 0.f; for (int i = 0; i < LI; ++i) ws += s_aggr[i];
    float wt = 0.f; for (int j = 0; j < LT; ++j) wt += s_txtw[j];
    s_sc[0] = ws; s_sc[1] = wt;
    float tmp[LI];
    for (int i = 0; i < LI; ++i) tmp[i] = s_aggr[i];
    for (int k = 0; k < 30; ++k) {
      int best = 0; float bv = tmp[0];
      for (int i = 1; i < LI; ++i) if (tmp[i] > bv) { bv = tmp[i]; best = i; }
      s_sel[best] = 1; tmp[best] = -3.4e38f;
    }
  }
  __syncthreads();

  // phase D: pooled outputs (channels t, t+256), data served from L2
  const float wsi = s_sc[0], wst = s_sc[1];
  float ik0 = 0.f, ik1 = 0.f, tm0 = 0.f, tm1 = 0.f;
  for (int i = 0; i < LI; ++i) {
    const float* r = &img[((size_t)i * B + b) * FEAT];
    float v0 = r[t], v1 = r[t + 256];
    float w = s_aggr[i] / wsi;
    tm0 += v0 * w; tm1 += v1 * w;
    if (s_sel[i]) { ik0 += v0; ik1 += v1; }
  }
  float tt0 = 0.f, tt1 = 0.f;
  for (int j = 0; j < LT; ++j) {
    if (!kpm[(size_t)b * LT + j]) {
      const float* r = &txt[((size_t)j * B + b) * FEAT];
      float w = s_txtw[j] / wst;
      tt0 += r[t] * w; tt1 += r[t + 256] * w;
    }
  }
  float* x0 = &X0[(size_t)b * DM];
  float* x1 = &X1[(size_t)b * DM];
  x0[t]              = ik0 / 30.0f;       x0[t + 256]       = ik1 / 30.0f;
  x0[FEAT + t]       = s_capmean[t];      x0[FEAT + t + 256] = s_capmean[t + 256];
  x1[t]              = tm0;               x1[t + 256]        = tm1;
  x1[FEAT + t]       = tt0;               x1[FEAT + t + 256] = tt1;
}

// ---------------------------------------------------------------------------
// x = LayerNorm(x + y) * g + b, D = 1024, one block (256 thr) per row
// ---------------------------------------------------------------------------
__global__ __launch_bounds__(256) void add_ln_kernel(
    float* __restrict__ X, const float* __restrict__ Y,
    const float* __restrict__ g, const float* __restrict__ be) {
  __shared__ float sbuf[8];
  const size_t row = (size_t)blockIdx.x * DM;
  const int t = threadIdx.x;
  float v[4];
  float s = 0.f;
#pragma unroll
  for (int k = 0; k < 4; ++k) {
    int d = t + k * 256;
    v[k] = X[row + d] + Y[row + d];
    s += v[k];
  }
  const float mean = block_sum256(s, sbuf) * (1.0f / DM);
  float s2 = 0.f;
#pragma unroll
  for (int k = 0; k < 4; ++k) { float c = v[k] - mean; s2 += c * c; }
  const float var = block_sum256(s2, sbuf) * (1.0f / DM);
  const float inv = rsqrtf(var + 1e-5f);
#pragma unroll
  for (int k = 0; k < 4; ++k) {
    int d = t + k * 256;
    X[row + d] = (v[k] - mean) * inv * g[d] + be[d];
  }
}

// ---------------------------------------------------------------------------
// row l2norm over D=1024 with split writes: outA=first 512, outB=second 512,
// outC (optional) = all 1024
// ---------------------------------------------------------------------------
__global__ __launch_bounds__(256) void l2norm_split_kernel(
    const float* __restrict__ X, float* __restrict__ outA,
    float* __restrict__ outB, float* __restrict__ outC) {
  __shared__ float sbuf[8];
  const size_t row = (size_t)blockIdx.x * DM;
  const int t = threadIdx.x;
  float v[4]; float ss = 0.f;
#pragma unroll
  for (int k = 0; k < 4; ++k) { v[k] = X[row + t + k * 256]; ss += v[k] * v[k]; }
  ss = block_sum256(ss, sbuf);
  const float inv = 1.0f / fmaxf(sqrtf(ss), 1e-12f);
  const size_t hrow = (size_t)blockIdx.x * FEAT;
#pragma unroll
  for (int k = 0; k < 4; ++k) {
    int d = t + k * 256;
    float y = v[k] * inv;
    if (d < FEAT) outA[hrow + d] = y; else outB[hrow + d - FEAT] = y;
    if (outC) outC[row + d] = y;
  }
}

// ---------------------------------------------------------------------------
// emb = l2norm(0.5*fu + 0.5*cls), D = 512
// ---------------------------------------------------------------------------
__global__ __launch_bounds__(256) void emb_mix_kernel(
    const float* __restrict__ fu, const float* __restrict__ cls,
    float* __restrict__ out) {
  __shared__ float sbuf[8];
  const size_t row = (size_t)blockIdx.x * FEAT;
  const int t = threadIdx.x;
  float v0 = 0.5f * fu[row + t]       + 0.5f * cls[row + t];
  float v1 = 0.5f * fu[row + t + 256] + 0.5f * cls[row + t + 256];
  float ss = block_sum256(v0 * v0 + v1 * v1, sbuf);
  const float inv = 1.0f / fmaxf(sqrtf(ss), 1e-12f);
  out[row + t] = v0 * inv;
  out[row + t + 256] = v1 * inv;
}

// ---------------------------------------------------------------------------
// row l2norm over D=64, one wave per row (lane owns 2 elems)
// ---------------------------------------------------------------------------
__global__ __launch_bounds__(32) void l2norm64_kernel(
    const float* __restrict__ X, float* __restrict__ out) {
  const size_t row = (size_t)blockIdx.x * NBITS;
  const int l = threadIdx.x;
  float v0 = X[row + l], v1 = X[row + l + 32];
  float ss = wave_sum(v0 * v0 + v1 * v1);
  const float inv = 1.0f / fmaxf(sqrtf(ss), 1e-12f);
  out[row + l] = v0 * inv;
  out[row + l + 32] = v1 * inv;
}

// ---------------------------------------------------------------------------
// host launch
// ---------------------------------------------------------------------------
struct LayerW {
  const float *ff1_b, *ff1_w, *ff2_b, *ff2_w, *in_b, *in_w;
  const float *ln1_b, *ln1_g, *ln2_b, *ln2_g, *out_b, *out_w;
};

static inline void gemm(const float* A, const float* W, const float* bias,
                        float* C, int M, int N, int K, int act,
                        hipStream_t s) {
  dim3 grid(N / TN, M / TM);
  gemm_nt_wmma_f32<<<grid, 256, 0, s>>>(A, W, bias, C, M, N, K, act);
}

extern "C" void kernel_launch(void* const* d_in, const int* in_sizes, int n_in,
                              void* d_out, int out_size, void* d_ws,
                              size_t ws_size, hipStream_t stream) {
  // -- inputs: setup_inputs() dict order --
  const float* img_tokens = (const float*)d_in[0];        // [50,2048,512]
  const float* txt_tokens = (const float*)d_in[1];        // [77,2048,512]
  const float* img_cls    = (const float*)d_in[2];        // [2048,512]
  const float* txt_eos    = (const float*)d_in[3];        // [2048,512]
  const unsigned char* kpm = (const unsigned char*)d_in[4]; // [2048,77] bool
  // -- params: jax pytree leaf order (dict keys sorted alphabetically) --
  int p = 5;
  const float* im_fc1_b  = (const float*)d_in[p++];
  const float* im_fc1_w  = (const float*)d_in[p++];
  const float* im_hash_b = (const float*)d_in[p++];
  const float* im_hash_w = (const float*)d_in[p++];
  LayerW lay[2];
  for (int l = 0; l < 2; ++l) {
    lay[l].ff1_b = (const float*)d_in[p++]; lay[l].ff1_w = (const float*)d_in[p++];
    lay[l].ff2_b = (const float*)d_in[p++]; lay[l].ff2_w = (const float*)d_in[p++];
    lay[l].in_b  = (const float*)d_in[p++]; lay[l].in_w  = (const float*)d_in[p++];
    lay[l].ln1_b = (const float*)d_in[p++]; lay[l].ln1_g = (const float*)d_in[p++];
    lay[l].ln2_b = (const float*)d_in[p++]; lay[l].ln2_g = (const float*)d_in[p++];
    lay[l].out_b = (const float*)d_in[p++]; lay[l].out_w = (const float*)d_in[p++];
  }
  const float* tx_fc1_b  = (const float*)d_in[p++];
  const float* tx_fc1_w  = (const float*)d_in[p++];
  const float* tx_hash_b = (const float*)d_in[p++];
  const float* tx_hash_w = (const float*)d_in[p++];

  // -- output slices (flat concat in reference return order) --
  float* out = (float*)d_out;
  float* o_img_fu     = out;                                   // [B,512]
  float* o_txt_fu     = o_img_fu + (size_t)B * FEAT;
  float* o_img_region = o_txt_fu + (size_t)B * FEAT;
  float* o_txt_region = o_img_region + (size_t)B * FEAT;
  float* o_all_region = o_txt_region + (size_t)B * FEAT;       // [B,1024]
  float* o_img_emb    = o_all_region + (size_t)B * DM;
  float* o_txt_emb    = o_img_emb + (size_t)B * FEAT;
  float* o_d_img      = o_txt_emb + (size_t)B * FEAT;          // [B,64]
  float* o_d_txt      = o_d_img + (size_t)B * NBITS;

  // -- workspace --
  float* ws = (float*)d_ws;
  float* X0 = ws;                              // [B,1024] fuse#1 state
  float* X1 = X0 + (size_t)B * DM;             // [B,1024] fuse#2 state
  float* T0 = X1 + (size_t)B * DM;             // [B,1024]
  float* T1 = T0 + (size_t)B * DM;             // [B,2048]
  float* T2 = T1 + (size_t)B * DFF;            // [B,4096]

  // 1) pooling: builds concat inputs for both fuse calls
  pool_kernel<<<B, 256, 0, stream>>>(img_tokens, txt_tokens, kpm, X0, X1);

  // 2) shared-weight fuse transformer (S=1 -> attention(x) == v-projection)
  auto run_fuse = [&](float* X) {
    for (int l = 0; l < 2; ++l) {
      const LayerW& L = lay[l];
      // v = x @ Wv^T + bv   (Wv = rows [2048,3072) of in_w)
      gemm(X, L.in_w + (size_t)2 * DM * DM, L.in_b + 2 * DM, T0, B, DM, DM, 0, stream);
      // attn_out = v @ out_w^T + out_b
      gemm(T0, L.out_w, L.out_b, T1, B, DM, DM, 0, stream);
      add_ln_kernel<<<B, 256, 0, stream>>>(X, T1, L.ln1_g, L.ln1_b);
      // ff
      gemm(X, L.ff1_w, L.ff1_b, T1, B, DFF, DM, 1, stream);
      gemm(T1, L.ff2_w, L.ff2_b, T0, B, DM, DFF, 0, stream);
      add_ln_kernel<<<B, 256, 0, stream>>>(X, T0, L.ln2_g, L.ln2_b);
    }
  };

  run_fuse(X0);
  l2norm_split_kernel<<<B, 256, 0, stream>>>(X0, o_img_region, o_txt_region,
                                             o_all_region);
  run_fuse(X1);
  l2norm_split_kernel<<<B, 256, 0, stream>>>(X1, o_img_fu, o_txt_fu, nullptr);

  // 3) embedding mix
  emb_mix_kernel<<<B, 256, 0, stream>>>(o_img_fu, img_cls, o_img_emb);
  emb_mix_kernel<<<B, 256, 0, stream>>>(o_txt_fu, txt_eos, o_txt_emb);

  // 4) hashing MLPs (input already unit-norm; _l2norm is idempotent)
  gemm(o_img_emb, im_fc1_w, im_fc1_b, T2, B, HID, FEAT, 1, stream);
  gemm(T2, im_hash_w, im_hash_b, T0, B, NBITS, HID, 2, stream);
  l2norm64_kernel<<<B, 32, 0, stream>>>(T0, o_d_img);

  gemm(o_txt_emb, tx_fc1_w, tx_fc1_b, T2, B, HID, FEAT, 1, stream);
  gemm(T2, tx_hash_w, tx_hash_b, T0, B, NBITS, HID, 2, stream);
  l2norm64_kernel<<<B, 32, 0, stream>>>(T0, o_d_txt);
}